// OrthogonalWightMLP_83519934038258
// MI455X (gfx1250) — compile-verified
//
#include <hip/hip_runtime.h>
#include <math.h>
#include <stdint.h>

// ---------------------------------------------------------------------------
// Householder orthogonal-weight builder for MI455X (gfx1250, wave32).
//
//   Qu = H(u_0)...H(u_{ci-1}), Qv = H(v_0)...H(v_{co-1}), out = Qu @ Qv[:ci,:]
//
// Blocked compact-WY: 64 reflectors = I - W Y^T, so the reflector chain is
// ~96 rank-64 GEMM updates. GEMMs run on V_WMMA_F32_16X16X4_F32 with tiles
// DMA'd into LDS by the Tensor Data Mover (tensor_load_to_lds, TENSORcnt,
// double-buffered). Working set (~63MB) is L2-resident (192MB L2) -> the run
// is matrix-pipe bound; HBM traffic is just inputs+output (~30MB @ 23.3TB/s).
// ---------------------------------------------------------------------------

typedef __attribute__((ext_vector_type(2))) float v2f;
typedef __attribute__((ext_vector_type(8))) float v8f;
typedef __attribute__((ext_vector_type(4))) unsigned int u32x4;
typedef __attribute__((ext_vector_type(8))) int i32x8;
typedef __attribute__((ext_vector_type(4))) int i32x4;

#define RB 64             // reflector block size (rank of each WY update)
#define BM 128            // workgroup tile M
#define BN 64             // workgroup tile N
#define BK 32             // K stage depth
#define KSPLIT 256        // split-K chunk for the skinny C = M*W GEMM
#define LDA_S (BK + 4)    // padded LDS stride 36 (TDM pad: 4dw every 32dw)
#define LDB_S (BN + 4)    // padded LDS stride 68 (TDM pad: 4dw every 64dw)
#define A_TILE (BM * LDA_S)            // 4608 floats
#define B_TILE (BN * LDA_S)            // 2304 floats (>= BK*LDB_S = 2176)
#define BUF_FLOATS (A_TILE + B_TILE)   // one double-buffer stage

// ------------------- Tensor Data Mover 2D tile -> LDS -----------------------
// D# per CDNA5 ISA: group0 = {flags, lds_addr,
// global_addr, type=2}; group1 = {data_size=4B, pad cfg, tensor/tile dims,
// dim0 stride}. tile_w floats per row, tile_h rows, row stride in elements.
// pad codes: interval 4 -> every 32 dwords, 5 -> every 64 dwords; amount 3 ->
// insert 4 dwords (turns 32-float rows into stride-36, 64-float into 68).
__device__ static inline void tdm_load_2d(uint32_t lds_byte_addr, const float* gptr,
                                          uint32_t tile_w, uint32_t tile_h,
                                          uint64_t row_stride_elems,
                                          uint32_t pad_interval_code,
                                          uint32_t pad_amount_code) {
  const uint64_t ga = (uint64_t)(uintptr_t)gptr;
  u32x4 g0;
  g0.x = 1u;                                     // count=1, user mode, no gather
  g0.y = lds_byte_addr;                          // D#[63:32]  lds_addr
  g0.z = (uint32_t)ga;                           // global_addr[31:0]
  g0.w = (uint32_t)((ga >> 32) & 0x01ffffffu) | (2u << 30);  // addr[56:32], type=2
  i32x8 g1;
  g1[0] = (int)((2u << 16) | (1u << 20) |        // data_size=4B, pad_enable
                (pad_interval_code << 22) | (pad_amount_code << 25));
  g1[1] = (int)((tile_w & 0xffffu) << 16);       // tensor_dim0[15:0] @ bit48
  g1[2] = (int)(((tile_w >> 16) & 0xffffu) | ((tile_h & 0xffffu) << 16));
  g1[3] = (int)(((tile_h >> 16) & 0xffffu) | ((tile_w & 0xffffu) << 16)); // tile_dim0
  g1[4] = (int)(tile_h & 0xffffu);               // tile_dim1 ; tile_dim2 = 0
  g1[5] = (int)(uint32_t)row_stride_elems;       // tensor_dim0_stride[31:0]
  g1[6] = (int)(uint32_t)((row_stride_elems >> 32) & 0xffffu);
  g1[7] = 0;
  const i32x4 z4 = {0, 0, 0, 0};
#if defined(__clang_major__) && __clang_major__ >= 23
  const i32x8 z8 = {0, 0, 0, 0, 0, 0, 0, 0};
  __builtin_amdgcn_tensor_load_to_lds(g0, g1, z4, z4, z8, 0);
#else
  __builtin_amdgcn_tensor_load_to_lds(g0, g1, z4, z4, 0);
#endif
}

// --------------------------- column normalization ---------------------------
__global__ __launch_bounds__(256) void normalize_cols_k(const float* __restrict__ X,
                                                        float* __restrict__ Y, int d) {
  int c = blockIdx.x * blockDim.x + threadIdx.x;
  if (c >= d) return;
  float s = 0.f;
  for (int r = 0; r < d; ++r) {
    float x = X[(size_t)r * d + c];
    s += x * x;
  }
  float rinv = rsqrtf(s);
  for (int r = 0; r < d; ++r)
    Y[(size_t)r * d + c] = X[(size_t)r * d + c] * rinv;
}

// --------------------- per-block W factor (compact WY) ----------------------
// w_i = 2*(v_i - W_{<i} (Y_{<i}^T v_i)); blocks independent -> 1 WG per block.
__global__ __launch_bounds__(256) void build_w_k(const float* __restrict__ Y,
                                                 float* __restrict__ W, int d) {
  const int c0 = blockIdx.x * RB;
  const int tid = threadIdx.x;
  const int jslot = tid >> 2, sub = tid & 3;
  const int rchunk = d >> 2;

  __shared__ float part[RB][4];
  __shared__ float tsh[RB];

  for (int i = 0; i < RB; ++i) {
    if (jslot < i) {  // t_j = <Y[:,c0+j], Y[:,c0+i]>, 4 threads per j
      const float* colj = Y + c0 + jslot;
      const float* coli = Y + c0 + i;
      float p = 0.f;
      for (int r = sub * rchunk; r < (sub + 1) * rchunk; ++r)
        p += colj[(size_t)r * d] * coli[(size_t)r * d];
      part[jslot][sub] = p;
    }
    __syncthreads();
    if (sub == 0 && jslot < i)
      tsh[jslot] = part[jslot][0] + part[jslot][1] + part[jslot][2] + part[jslot][3];
    __syncthreads();
    for (int r = tid; r < d; r += 256) {
      const float* wrow = W + (size_t)r * d + c0;
      float u = 0.f;
      for (int j = 0; j < i; ++j) u += wrow[j] * tsh[j];
      W[(size_t)r * d + c0 + i] = 2.f * (Y[(size_t)r * d + c0 + i] - u);
    }
    __syncthreads();
  }
}

// ------------------------------- M = I --------------------------------------
__global__ __launch_bounds__(256) void set_identity_k(float* __restrict__ M, int d) {
  size_t idx = (size_t)blockIdx.x * blockDim.x + threadIdx.x;
  if (idx >= (size_t)d * d) return;
  int r = (int)(idx / (size_t)d), c = (int)(idx % (size_t)d);
  M[idx] = (r == c) ? 1.f : 0.f;
}

// ----------------- f32 WMMA GEMM: C = a*A*op(B) + b*Cin ---------------------
// A: MxK row-major. B: KxN row-major, or NxK accessed transposed (TRANSB).
// Tiles DMA'd by the TDM into double-buffered LDS; 8 waves/WG, each a 32x32
// tile of 2x2 v_wmma_f32_16x16x4_f32 accumulators. TRANSB stores Bs[j][k]
// (direct TDM copy, b64 fragment loads); NT stores Bs[k][j].
// SPLITK: blockIdx.z covers K chunks of KSPLIT; results atomicAdd'ed.
template <bool TRANSB, bool SPLITK>
__global__ __launch_bounds__(256) void gemm_wmma_f32(
    const float* __restrict__ A, int lda,
    const float* __restrict__ B, int ldb,
    const float* __restrict__ Cin, float* __restrict__ Cout, int ldc,
    int K, float alpha, float beta) {
  __shared__ float smem[2 * BUF_FLOATS];

  const int tid = threadIdx.x;
  const int lane = tid & 31;
  const int wave = tid >> 5;
  const int wm = wave & 3, wn = wave >> 2;
  const int half = lane >> 4, l = lane & 15;
  const int i0 = blockIdx.y * BM;
  const int j0 = blockIdx.x * BN;
  const bool issuer = (wave == 0);
  const uint32_t lds_base = (uint32_t)(uintptr_t)(&smem[0]);  // flat->LDS offset

  int kb = 0, ke = K;
  if (SPLITK) {
    kb = blockIdx.z * KSPLIT;
    ke = kb + KSPLIT;
    if (ke > K) ke = K;
  }

  v8f acc[2][2] = {};

  auto issue = [&](int k0, int p) {
    const uint32_t abase = lds_base + (uint32_t)(p * BUF_FLOATS) * 4u;
    const uint32_t bbase = abase + (uint32_t)A_TILE * 4u;
    // A tile: BM rows x BK floats -> LDS stride 36 via pad(32dw,4dw)
    tdm_load_2d(abase, A + (size_t)i0 * lda + k0, BK, BM, (uint64_t)lda, 4u, 3u);
    if (TRANSB)  // Bs[j][k]: BN rows x BK floats (natural copy of B^T block)
      tdm_load_2d(bbase, B + (size_t)j0 * ldb + k0, BK, BN, (uint64_t)ldb, 4u, 3u);
    else         // Bs[k][j]: BK rows x BN floats -> stride 68 via pad(64dw,4dw)
      tdm_load_2d(bbase, B + (size_t)k0 * ldb + j0, BN, BK, (uint64_t)ldb, 5u, 3u);
  };

  if (issuer) issue(kb, 0);

  int p = 0;
  for (int k0 = kb; k0 < ke; k0 += BK, p ^= 1) {
    const bool more = (k0 + BK) < ke;
    if (issuer) {
      if (more) {
        issue(k0 + BK, p ^ 1);                 // next stage in flight
        __builtin_amdgcn_s_wait_tensorcnt(2);  // current stage's 2 TDMs done
      } else {
        __builtin_amdgcn_s_wait_tensorcnt(0);
      }
    }
    __syncthreads();

    const float* Asp = smem + p * BUF_FLOATS;
    const float* Bsp = Asp + A_TILE;

#pragma unroll
    for (int kk = 0; kk < BK; kk += 4) {
      const int ka = kk + half * 2;  // lanes 0-15: K={kk,kk+1}; 16-31: {kk+2,kk+3}
      v2f a[2], b[2];
#pragma unroll
      for (int ti = 0; ti < 2; ++ti) {
        const float* ap = &Asp[(wm * 32 + ti * 16 + l) * LDA_S + ka];
        a[ti].x = ap[0];
        a[ti].y = ap[1];
      }
#pragma unroll
      for (int tj = 0; tj < 2; ++tj) {
        const int col = wn * 32 + tj * 16 + l;
        if (TRANSB) {
          const float* bp = &Bsp[col * LDA_S + ka];
          b[tj].x = bp[0];
          b[tj].y = bp[1];
        } else {
          b[tj].x = Bsp[(ka + 0) * LDB_S + col];
          b[tj].y = Bsp[(ka + 1) * LDB_S + col];
        }
      }
#pragma unroll
      for (int ti = 0; ti < 2; ++ti)
#pragma unroll
        for (int tj = 0; tj < 2; ++tj)
          acc[ti][tj] = __builtin_amdgcn_wmma_f32_16x16x4_f32(
              false, a[ti], false, b[tj], (short)0, acc[ti][tj], false, false);
    }
    __syncthreads();
  }

  // ---- epilogue: C/D layout -> VGPR r holds (M = r + 8*half, N = l) ----
#pragma unroll
  for (int ti = 0; ti < 2; ++ti)
#pragma unroll
    for (int tj = 0; tj < 2; ++tj) {
      const int col = j0 + wn * 32 + tj * 16 + l;
#pragma unroll
      for (int r = 0; r < 8; ++r) {
        const int row = i0 + wm * 32 + ti * 16 + half * 8 + r;
        float val = alpha * acc[ti][tj][r];
        if (SPLITK) {
          atomicAdd(&Cout[(size_t)row * ldc + col], val);
        } else {
          if (beta != 0.f) val += beta * Cin[(size_t)row * ldc + col];
          Cout[(size_t)row * ldc + col] = val;
        }
      }
    }
}

// ------------------------------- launcher -----------------------------------
extern "C" void kernel_launch(void* const* d_in, const int* in_sizes, int n_in,
                              void* d_out, int out_size, void* d_ws, size_t ws_size,
                              hipStream_t stream) {
  const float* U = (const float*)d_in[0];
  const float* V = (const float*)d_in[1];
  float* out = (float*)d_out;

  const int ci = (int)(sqrt((double)in_sizes[0]) + 0.5);  // 1024
  const int co = (int)(sqrt((double)in_sizes[1]) + 0.5);  // 2048
  const size_t ci2 = (size_t)ci * ci, co2 = (size_t)co * co;

  float* ws = (float*)d_ws;
  float* Yu = ws;
  float* Wu = Yu + ci2;
  float* Mu = Wu + ci2;
  float* Yv = Mu + ci2;
  float* Wv = Yv + co2;
  float* Mv = Wv + co2;
  float* Cb = Mv + co2;  // (d x RB) split-K accumulator for GEMM1

  normalize_cols_k<<<dim3((ci + 255) / 256), dim3(256), 0, stream>>>(U, Yu, ci);
  normalize_cols_k<<<dim3((co + 255) / 256), dim3(256), 0, stream>>>(V, Yv, co);
  build_w_k<<<dim3(ci / RB), dim3(256), 0, stream>>>(Yu, Wu, ci);
  build_w_k<<<dim3(co / RB), dim3(256), 0, stream>>>(Yv, Wv, co);
  set_identity_k<<<dim3((unsigned)((ci2 + 255) / 256)), dim3(256), 0, stream>>>(Mu, ci);
  set_identity_k<<<dim3((unsigned)((co2 + 255) / 256)), dim3(256), 0, stream>>>(Mv, co);

  // Qu blocks: M <- M - (M W_b) Y_b^T
  for (int b = 0; b < ci / RB; ++b) {
    const int c0 = b * RB;
    hipMemsetAsync(Cb, 0, (size_t)ci * RB * sizeof(float), stream);
    gemm_wmma_f32<false, true><<<dim3(RB / BN, ci / BM, ci / KSPLIT), 256, 0, stream>>>(
        Mu, ci, Wu + c0, ci, nullptr, Cb, RB, ci, 1.f, 0.f);
    gemm_wmma_f32<true, false><<<dim3(ci / BN, ci / BM), 256, 0, stream>>>(
        Cb, RB, Yu + c0, ci, Mu, Mu, ci, RB, -1.f, 1.f);
  }
  // Qv blocks
  for (int b = 0; b < co / RB; ++b) {
    const int c0 = b * RB;
    hipMemsetAsync(Cb, 0, (size_t)co * RB * sizeof(float), stream);
    gemm_wmma_f32<false, true><<<dim3(RB / BN, co / BM, co / KSPLIT), 256, 0, stream>>>(
        Mv, co, Wv + c0, co, nullptr, Cb, RB, co, 1.f, 0.f);
    gemm_wmma_f32<true, false><<<dim3(co / BN, co / BM), 256, 0, stream>>>(
        Cb, RB, Yv + c0, co, Mv, Mv, co, RB, -1.f, 1.f);
  }
  // weight = Qu (ci x ci) @ Qv[:ci, :] (ci x co)
  gemm_wmma_f32<false, false><<<dim3(co / BN, ci / BM), 256, 0, stream>>>(
      Mu, ci, Mv, co, nullptr, out, co, ci, 1.f, 0.f);
}